// SpatialTransformer_57157424775195
// MI455X (gfx1250) — compile-verified
//
#include <hip/hip_runtime.h>
#include <hip/hip_bf16.h>
#include <stdint.h>

// MI455X (gfx1250) 3D spatial transformer (grid_sample, align_corners=True,
// padding_mode='zeros'). Memory-bound gather: ~295 MB compulsory traffic ->
// ~12.7 us floor at 23.3 TB/s. No matmul content -> no WMMA; the CDNA5 path
// that applies is the Tensor Data Mover: flow is staged per-block into LDS as
// a 3x256 2D tile via tensor_load_to_lds (TENSORcnt), and output uses
// non-temporal stores so the 192MB L2 stays dedicated to src (98MB, fits).

typedef unsigned int v4u __attribute__((ext_vector_type(4)));
typedef int          v8i __attribute__((ext_vector_type(8)));
typedef int          v4i __attribute__((ext_vector_type(4)));

#define SDIM   160u
#define DHW    (160u * 160u * 160u)   /* 4,096,000 */
#define CHUNKS (DHW / 256u)           /* 16,000    */
#define NBATCH 2u
#define NCHAN  3u

__global__ __launch_bounds__(256) void st3d_kernel(
    const float* __restrict__ src, const float* __restrict__ flow,
    const int* __restrict__ is_msk, float* __restrict__ out)
{
  __shared__ float sflow[3 * 256];

  const unsigned tid   = threadIdx.x;
  const unsigned chunk = blockIdx.x % CHUNKS;
  const unsigned b     = blockIdx.x / CHUNKS;
  const unsigned p0    = chunk * 256u;

  // ---- TDM: stage flow[b, 0:3, p0:p0+256] into LDS as a 3-row 2D tile ----
  if (tid < 32u) {  // wave 0 issues the DMA (TDM ignores EXEC; one issue/wave)
    const uint64_t ga  = (uint64_t)(uintptr_t)(const void*)
                         (flow + (size_t)b * 3u * DHW + p0);
    const unsigned lds = (unsigned)(uintptr_t)(void*)&sflow[0]; // addr[31:0] == LDS byte offset

    v4u g0;
    g0.x = 1u;                                  // count=1 valid descriptor
    g0.y = lds;                                 // lds_addr (bytes)
    g0.z = (unsigned)(ga & 0xFFFFFFFFull);      // global_addr[31:0]
    g0.w = (unsigned)(ga >> 32) | (2u << 30);   // global_addr[56:32] | type=2

    v8i g1;
    g1[0] = 0x00020000;          // workgroup_mask=0, data_size=2 (4 bytes)
    g1[1] = (int)0x80000000u;    // tensor_dim0[15:0]=0x8000 (DHW=0x3E8000) in [31:16]
    g1[2] = 0x0003003E;          // tensor_dim0[31:16]=0x3E, tensor_dim1=3
    g1[3] = (int)(256u << 16);   // tile_dim0 = 256
    g1[4] = 3;                   // tile_dim1 = 3, tile_dim2 = 0
    g1[5] = (int)DHW;            // tensor_dim0_stride[31:0] = DHW
    g1[6] = 0;                   // stride[47:32]=0, tensor_dim1_stride lo=0
    g1[7] = 0;

    v4i gz4 = {0, 0, 0, 0};              // groups 2/3 unused (2D tile)
    v8i gz8 = {0, 0, 0, 0, 0, 0, 0, 0};  // extra operand of 6-arg builtin
    __builtin_amdgcn_tensor_load_to_lds(g0, g1, gz4, gz4, gz8, 0);
    __builtin_amdgcn_s_wait_tensorcnt(0);
  }
  __syncthreads();

  // ---- per-point sampling ----
  const unsigned p  = p0 + tid;
  const unsigned x  = p % SDIM;
  const unsigned t1 = p / SDIM;
  const unsigned y  = t1 % SDIM;
  const unsigned z  = t1 / SDIM;

  const float fz = (float)z + sflow[tid];          // flow ch0 -> z
  const float fy = (float)y + sflow[256u + tid];   // flow ch1 -> y
  const float fx = (float)x + sflow[512u + tid];   // flow ch2 -> x

  const float* __restrict__ sb = src + (size_t)b * 3u * DHW;
  float*       __restrict__ ob = out + (size_t)b * 3u * DHW + p;

  if (*is_msk == 0) {
    // trilinear, zero padding: fold per-axis validity into weights
    const float z0f = floorf(fz), y0f = floorf(fy), x0f = floorf(fx);
    const float tz = fz - z0f, ty = fy - y0f, tx = fx - x0f;
    const int z0 = (int)z0f, y0i = (int)y0f, x0i = (int)x0f;
    const int z1 = z0 + 1,  y1  = y0i + 1,  x1  = x0i + 1;

    const float wz0 = ((unsigned)z0  < SDIM) ? (1.f - tz) : 0.f;
    const float wz1 = ((unsigned)z1  < SDIM) ? tz         : 0.f;
    const float wy0 = ((unsigned)y0i < SDIM) ? (1.f - ty) : 0.f;
    const float wy1 = ((unsigned)y1  < SDIM) ? ty         : 0.f;
    const float wx0 = ((unsigned)x0i < SDIM) ? (1.f - tx) : 0.f;
    const float wx1 = ((unsigned)x1  < SDIM) ? tx         : 0.f;

    const unsigned cz0 = (unsigned)min(max(z0, 0), 159);
    const unsigned cz1 = (unsigned)min(max(z1, 0), 159);
    const unsigned cy0 = (unsigned)min(max(y0i, 0), 159);
    const unsigned cy1 = (unsigned)min(max(y1, 0), 159);
    const unsigned cx0 = (unsigned)min(max(x0i, 0), 159);
    const unsigned cx1 = (unsigned)min(max(x1, 0), 159);

    const unsigned oz0 = cz0 * (SDIM * SDIM), oz1 = cz1 * (SDIM * SDIM);
    const unsigned oy0 = cy0 * SDIM,          oy1 = cy1 * SDIM;

    const unsigned i000 = oz0 + oy0 + cx0, i001 = oz0 + oy0 + cx1;
    const unsigned i010 = oz0 + oy1 + cx0, i011 = oz0 + oy1 + cx1;
    const unsigned i100 = oz1 + oy0 + cx0, i101 = oz1 + oy0 + cx1;
    const unsigned i110 = oz1 + oy1 + cx0, i111 = oz1 + oy1 + cx1;

    const float w000 = wz0 * wy0 * wx0, w001 = wz0 * wy0 * wx1;
    const float w010 = wz0 * wy1 * wx0, w011 = wz0 * wy1 * wx1;
    const float w100 = wz1 * wy0 * wx0, w101 = wz1 * wy0 * wx1;
    const float w110 = wz1 * wy1 * wx0, w111 = wz1 * wy1 * wx1;

#pragma unroll
    for (unsigned c = 0; c < NCHAN; ++c) {
      const float* __restrict__ sc = sb + c * DHW;
      float a = w000 * sc[i000];
      a = fmaf(w001, sc[i001], a);
      a = fmaf(w010, sc[i010], a);
      a = fmaf(w011, sc[i011], a);
      a = fmaf(w100, sc[i100], a);
      a = fmaf(w101, sc[i101], a);
      a = fmaf(w110, sc[i110], a);
      a = fmaf(w111, sc[i111], a);
      __builtin_nontemporal_store(a, ob + c * DHW);
    }
  } else {
    // nearest: round-half-to-even (rintf in default RNE mode == jnp.round)
    const int iz = (int)rintf(fz), iy = (int)rintf(fy), ix = (int)rintf(fx);
    const float m = ((unsigned)iz < SDIM && (unsigned)iy < SDIM &&
                     (unsigned)ix < SDIM) ? 1.f : 0.f;
    const unsigned ci = ((unsigned)min(max(iz, 0), 159) * SDIM +
                         (unsigned)min(max(iy, 0), 159)) * SDIM +
                         (unsigned)min(max(ix, 0), 159);
#pragma unroll
    for (unsigned c = 0; c < NCHAN; ++c)
      __builtin_nontemporal_store(m * sb[c * DHW + ci], ob + c * DHW);
  }
}

extern "C" void kernel_launch(void* const* d_in, const int* in_sizes, int n_in,
                              void* d_out, int out_size, void* d_ws, size_t ws_size,
                              hipStream_t stream) {
  const float* src    = (const float*)d_in[0];
  const float* flow   = (const float*)d_in[1];
  const int*   is_msk = (const int*)d_in[2];
  float*       out    = (float*)d_out;
  (void)in_sizes; (void)n_in; (void)out_size; (void)d_ws; (void)ws_size;

  dim3 grid(NBATCH * CHUNKS);  // 32,000 blocks
  dim3 block(256);             // 8 wave32s
  st3d_kernel<<<grid, block, 0, stream>>>(src, flow, is_msk, out);
}